// SpatialTransformer2dAffineLayer_86663850098768
// MI455X (gfx1250) — compile-verified
//
#include <hip/hip_runtime.h>
#include <math.h>

// ---------------------------------------------------------------------------
// Spatial Transformer (affine) for MI455X / gfx1250.
//
// Stage 1: theta = tanh(theta_feat @ W + b)   (16x128)x(128x6)
//   -> one wave32, V_WMMA_F32_16X16X4_F32 over K=128 (32 WMMA issues),
//      B padded 6->16 columns with zeros. Exact f32 math.
//
// Stage 2: bilinear warp (16,224,224,32) f32 -> (16,224,224,32) f32.
//   Pure bandwidth (~206 MB traffic vs ~180 MFLOP  => HBM-bound, ~9us at
//   23.3 TB/s). Each lane handles 4 channels (native float4), 8 lanes per
//   pixel -> every gather is a coalesced global_load_b128. Output is written
//   with non-temporal (NT) stores: the 103 MB input fits in the 192 MB L2
//   and is re-read ~4x by overlapping bilinear footprints, while the output
//   is write-once; NT stores keep the output stream from evicting the input
//   working set.
// ---------------------------------------------------------------------------

typedef __attribute__((ext_vector_type(2))) float v2f;
typedef __attribute__((ext_vector_type(4))) float v4f;
typedef __attribute__((ext_vector_type(8))) float v8f;

// ---------------- Stage 1: WMMA GEMM (16x128 @ 128x6) + bias + tanh --------
__global__ __launch_bounds__(32) void stn_theta_wmma(
    const float* __restrict__ feat,   // (16,128) row-major
    const float* __restrict__ Wm,     // (128,6)  row-major
    const float* __restrict__ bias,   // (6)
    float* __restrict__ theta)        // (16,6)   row-major, in workspace
{
    const int lane = threadIdx.x;     // 0..31, wave32
    const int m    = lane & 15;       // A row / B column index
    const int hi   = lane >> 4;       // half-wave select

    v8f acc = {};

    // A 16x4 f32 layout: VGPR0 = K=0 (lanes 0-15) / K=2 (lanes 16-31),
    //                    VGPR1 = K=1 / K=3. B mirrors across half-waves.
    for (int k = 0; k < 128; k += 4) {
        // A tile: row m, K = k + 2*hi + {0,1}; contiguous 8B-aligned pair
        const float* ap = feat + m * 128 + k + hi * 2;
        v2f a;
        a.x = ap[0];
        a.y = ap[1];

        // B tile: col n = m (zero for n >= 6), K = k + 2*hi + {0,1}
        const int   n   = m;
        const float msk = (n < 6) ? 1.0f : 0.0f;
        const int   ncl = (n < 6) ? n : 5;        // clamp keeps loads in-bounds
        v2f bm;
        bm.x = Wm[(k + hi * 2    ) * 6 + ncl] * msk;
        bm.y = Wm[(k + hi * 2 + 1) * 6 + ncl] * msk;

        // 8 args: (neg_a, A, neg_b, B, c_mod, C, reuse_a, reuse_b)
        acc = __builtin_amdgcn_wmma_f32_16x16x4_f32(
            false, a, false, bm, (short)0, acc, false, false);
    }

    // C/D layout: VGPR r -> (M=r, N=lane) for lanes 0-15, (M=r+8, N=lane-16)
    const int N = m;
    if (N < 6) {
        const float bb = bias[N];
#pragma unroll
        for (int r = 0; r < 8; ++r) {
            const int M = hi ? (r + 8) : r;
            theta[M * 6 + N] = tanhf(acc[r] + bb);
        }
    }
}

// ---------------- Stage 2: bilinear sampling (bandwidth-bound) -------------
__global__ __launch_bounds__(256) void stn_sample(
    const float* __restrict__ x,      // (16,224,224,32)
    const float* __restrict__ theta,  // (16,6)
    float* __restrict__ out)          // (16,224,224,32)
{
    const int tid = blockIdx.x * 256 + threadIdx.x;
    const int c4  = (tid & 7) * 4;    // channel group: 4 floats per lane
    const int p   = tid >> 3;         // pixel index 0..802815

    const int j = p % 224;            // output x
    const int t = p / 224;
    const int i = t % 224;            // output y
    const int b = t / 224;            // batch

    const float* th = theta + b * 6;
    const float xt = -1.0f + (2.0f / 223.0f) * (float)j;
    const float yt = -1.0f + (2.0f / 223.0f) * (float)i;

    const float xs = th[0] * xt + th[1] * yt + th[2];
    const float ys = th[3] * xt + th[4] * yt + th[5];

    const float xp = (xs + 1.0f) * 112.0f;
    const float yp = (ys + 1.0f) * 112.0f;

    int x0 = (int)floorf(xp);
    int y0 = (int)floorf(yp);
    int x1 = x0 + 1;
    int y1 = y0 + 1;
    x0 = min(max(x0, 0), 223);
    x1 = min(max(x1, 0), 223);
    y0 = min(max(y0, 0), 223);
    y1 = min(max(y1, 0), 223);

    const float x0f = (float)x0, x1f = (float)x1;
    const float y0f = (float)y0, y1f = (float)y1;
    const float wa = (x1f - xp) * (y1f - yp);
    const float wb = (x1f - xp) * (yp - y0f);
    const float wc = (xp - x0f) * (y1f - yp);
    const float wd = (xp - x0f) * (yp - y0f);

    const float* base = x + (size_t)b * (224 * 224 * 32);
    const v4f Ia = *(const v4f*)(base + ((y0 * 224 + x0) * 32 + c4));
    const v4f Ib = *(const v4f*)(base + ((y1 * 224 + x0) * 32 + c4));
    const v4f Ic = *(const v4f*)(base + ((y0 * 224 + x1) * 32 + c4));
    const v4f Id = *(const v4f*)(base + ((y1 * 224 + x1) * 32 + c4));

    const v4f r = wa * Ia + wb * Ib + wc * Ic + wd * Id;

    // Write-once output: non-temporal store keeps the reusable input
    // working set resident in the 192 MB L2.
    __builtin_nontemporal_store(r, (v4f*)(out + (size_t)tid * 4));
}

// ---------------------------------------------------------------------------
extern "C" void kernel_launch(void* const* d_in, const int* in_sizes, int n_in,
                              void* d_out, int out_size, void* d_ws, size_t ws_size,
                              hipStream_t stream) {
    const float* x    = (const float*)d_in[0];  // (16,224,224,32)
    const float* feat = (const float*)d_in[1];  // (16,128)
    const float* Wm   = (const float*)d_in[2];  // (128,6)
    const float* bias = (const float*)d_in[3];  // (6)
    float* out   = (float*)d_out;
    float* theta = (float*)d_ws;                // 96 floats

    stn_theta_wmma<<<1, 32, 0, stream>>>(feat, Wm, bias, theta);

    // 16*224*224 pixels * 8 threads/pixel = 6,422,528 threads = 25,088 blocks
    stn_sample<<<25088, 256, 0, stream>>>(x, theta, out);
}